// TimeAttn_21474836480623
// MI455X (gfx1250) — compile-verified
//
#include <hip/hip_runtime.h>
#include <hip/hip_bf16.h>
#include <math.h>

#define DIM   256
#define LSEQ  128
#define TDIM  64
#define VOCAB 150
#define NB    2048

typedef float v2f __attribute__((ext_vector_type(2)));
typedef float v8f __attribute__((ext_vector_type(8)));
typedef int   v4i __attribute__((ext_vector_type(4)));

#if defined(__has_builtin)
#if __has_builtin(__builtin_amdgcn_global_load_async_to_lds_b128)
#define HAVE_ASYNC_LDS 1
#endif
#endif

#if HAVE_ASYNC_LDS
typedef __attribute__((address_space(1))) v4i* gptr4_t;
typedef __attribute__((address_space(3))) v4i* lptr4_t;
#endif

// ---------------------------------------------------------------------------
// GEMM C[M=?,256] = A[M,256] (x) B, K = 256, using V_WMMA_F32_16X16X4_F32.
// BT==1: C[m,n] = sum_k A[m,k] * Bm[n,k]   (B transposed, i.e. y = x @ W^T)
// BT==0: C[m,n] = sum_k A[m,k] * Bm[k,n]
// One wave (32 threads) computes one 16x16 tile of C.
// ---------------------------------------------------------------------------
template <int BT>
__global__ __launch_bounds__(32)
void gemm256_wmma(const float* __restrict__ A,
                  const float* __restrict__ Bm,
                  float* __restrict__ C)
{
    const int m0   = blockIdx.x * 16;
    const int n0   = blockIdx.y * 16;
    const int lane = threadIdx.x;       // 0..31
    const int half = lane >> 4;         // 0/1
    const int lm   = lane & 15;

    v8f acc = {};
    for (int k0 = 0; k0 < 256; k0 += 4) {
        const int ka = k0 + 2 * half;
        // A fragment: lane holds A[m0+lm, ka], A[m0+lm, ka+1]
        v2f afrag = *(const v2f*)(A + (size_t)(m0 + lm) * 256 + ka);
        v2f bfrag;
        if (BT) {
            // B[k,n] = Bm[n,k] -> lane holds Bm[n0+lm, ka..ka+1]
            bfrag = *(const v2f*)(Bm + (size_t)(n0 + lm) * 256 + ka);
        } else {
            // B[k,n] = Bm[k,n] -> lane holds Bm[ka, n0+lm], Bm[ka+1, n0+lm]
            bfrag.x = Bm[(size_t)(ka + 0) * 256 + n0 + lm];
            bfrag.y = Bm[(size_t)(ka + 1) * 256 + n0 + lm];
        }
        acc = __builtin_amdgcn_wmma_f32_16x16x4_f32(
            /*neg_a=*/false, afrag, /*neg_b=*/false, bfrag,
            /*c_mod=*/(short)0, acc, /*reuse_a=*/false, /*reuse_b=*/false);
    }

    // C/D layout: VGPR i -> row (i + 8*half), col = lm
    #pragma unroll
    for (int i = 0; i < 8; ++i) {
        C[(size_t)(m0 + i + 8 * half) * 256 + n0 + lm] = acc[i];
    }
}

// ---------------------------------------------------------------------------
// Fused attention: one workgroup per batch row b.
//  LDS <- out[b] (128 KB, async-to-LDS), dot with q~[b], + embedding term,
//  masked softmax over L, weighted sum back to y[b, :].
// ---------------------------------------------------------------------------
__global__ __launch_bounds__(256)
void attn_fused(const float* __restrict__ out,
                const float* __restrict__ qt,
                const int*   __restrict__ pos_ind,
                const int*   __restrict__ rel_dt,
                const float* __restrict__ pos_emb,
                const float* __restrict__ rel_emb,
                const float* __restrict__ Wrp,
                float* __restrict__ y)
{
    extern __shared__ float smem[];
    float* out_s  = smem;                    // LSEQ*DIM = 32768 floats
    float* attn_s = smem + LSEQ * DIM;       // 128 floats
    float* red_s  = attn_s + LSEQ;           // scratch

    const int b    = blockIdx.x;
    const int t    = threadIdx.x;            // 0..255
    const int lane = t & 31;
    const int wave = t >> 5;                 // 0..7

    const float* outb = out + (size_t)b * (LSEQ * DIM);

    // ---- Phase 0: stage floats 0..8191 of out[b] into LDS (2048 float4 chunks, 8 per thread)
#if HAVE_ASYNC_LDS
    #pragma unroll
    for (int i = 0; i < 8; ++i) {
        const int idx = (t + i * 256) * 4;   // float index, 16B aligned
        __builtin_amdgcn_global_load_async_to_lds_b128(
            (gptr4_t)(outb + idx),
            (lptr4_t)(out_s + idx),
            0, 0);
    }
#if __has_builtin(__builtin_amdgcn_s_wait_asynccnt)
    __builtin_amdgcn_s_wait_asynccnt(0);
#else
    asm volatile("s_wait_asynccnt 0" ::: "memory");
#endif
#else
    #pragma unroll
    for (int i = 0; i < 8; ++i) {
        const int idx = (t + i * 256) * 4;
        *(float4*)(out_s + idx) = *(const float4*)(outb + idx);
    }
#endif
    __syncthreads();

    // ---- Phase 1: attn_0[l] = out[b,l,:] . q~[b,:]  (one wave per 16 rows)
    const float4 q0 = *(const float4*)(qt + (size_t)b * DIM + lane * 8);
    const float4 q1 = *(const float4*)(qt + (size_t)b * DIM + lane * 8 + 4);
    for (int l = wave * 16; l < wave * 16 + 16; ++l) {
        const float* row = out_s + l * DIM + lane * 8;
        const float4 x0 = *(const float4*)(row);
        const float4 x1 = *(const float4*)(row + 4);
        float d = x0.x * q0.x + x0.y * q0.y + x0.z * q0.z + x0.w * q0.w
                + x1.x * q1.x + x1.y * q1.y + x1.z * q1.z + x1.w * q1.w;
        #pragma unroll
        for (int m = 16; m; m >>= 1) d += __shfl_xor(d, m, 32);
        if (lane == 0) attn_s[l] = d;
    }
    __syncthreads();

    // ---- Phase 1b: + BETA * (concat(rel_e,pos_e) . Wrp), pad mask, /temp
    if (t < LSEQ) {
        const int pi = pos_ind[(size_t)b * LSEQ + t];
        const int rd = rel_dt[(size_t)b * LSEQ + t];
        const float4* rr = (const float4*)(rel_emb + (size_t)rd * TDIM);
        const float4* pr = (const float4*)(pos_emb + (size_t)pi * TDIM);
        const float4* wr = (const float4*)(Wrp);     // 2*TDIM = 128 floats
        float a1 = 0.0f;
        #pragma unroll
        for (int i = 0; i < TDIM / 4; ++i) {
            const float4 rv = rr[i], wv = wr[i];
            a1 += rv.x * wv.x + rv.y * wv.y + rv.z * wv.z + rv.w * wv.w;
        }
        #pragma unroll
        for (int i = 0; i < TDIM / 4; ++i) {
            const float4 pv = pr[i], wv = wr[TDIM / 4 + i];
            a1 += pv.x * wv.x + pv.y * wv.y + pv.z * wv.z + pv.w * wv.w;
        }
        // temperature = sqrt(256) = 16; BETA = 0.5
        const float a = attn_s[t] * 0.0625f + 0.5f * a1;
        attn_s[t] = (pi == 0) ? -INFINITY : a;
    }
    __syncthreads();

    // ---- Phase 2: softmax over L (wave 0 reduces)
    if (wave == 0) {
        float m = fmaxf(fmaxf(attn_s[lane], attn_s[lane + 32]),
                        fmaxf(attn_s[lane + 64], attn_s[lane + 96]));
        #pragma unroll
        for (int s = 16; s; s >>= 1) m = fmaxf(m, __shfl_xor(m, s, 32));
        red_s[0] = m;
    }
    __syncthreads();
    const float mx = red_s[0];
    if (t < LSEQ) attn_s[t] = __expf(attn_s[t] - mx);
    __syncthreads();
    if (wave == 0) {
        float s = attn_s[lane] + attn_s[lane + 32]
                + attn_s[lane + 64] + attn_s[lane + 96];
        #pragma unroll
        for (int k = 16; k; k >>= 1) s += __shfl_xor(s, k, 32);
        red_s[1] = 1.0f / s;
    }
    __syncthreads();
    const float inv = red_s[1];

    // ---- Phase 3: y[b,d] = inv * sum_l attn[l] * out[b,l,d]  (d = t)
    float acc = 0.0f;
    #pragma unroll 4
    for (int l = 0; l < LSEQ; ++l) {
        acc += attn_s[l] * out_s[l * DIM + t];
    }
    y[(size_t)b * DIM + t] = acc * inv;
}

// ---------------------------------------------------------------------------
extern "C" void kernel_launch(void* const* d_in, const int* in_sizes, int n_in,
                              void* d_out, int out_size, void* d_ws, size_t ws_size,
                              hipStream_t stream) {
    const float* out_t   = (const float*)d_in[0];  // [B, L, D]
    const float* hn      = (const float*)d_in[1];  // [B, D]
    const int*   pos_ind = (const int*)  d_in[2];  // [B, L]
    const int*   rel_dt  = (const int*)  d_in[3];  // [B, L]
    /* d_in[4] abs_dt: unused by reference */
    const float* pos_emb = (const float*)d_in[5];  // [VOCAB, T]
    const float* rel_emb = (const float*)d_in[6];  // [VOCAB, T]
    const float* Wk      = (const float*)d_in[7];  // [D, D]
    const float* Wq      = (const float*)d_in[8];  // [D, D]
    const float* Wrp     = (const float*)d_in[9];  // [1, 2T]
    float* y = (float*)d_out;                      // [B, D]

    float* tmp = (float*)d_ws;                     // attn_q: [B, D]
    float* qt  = tmp + (size_t)NB * DIM;           // q~    : [B, D]

    // q~[b] = Wk^T @ (Wq @ hn[b]) : two small WMMA GEMMs
    dim3 gg(NB / 16, DIM / 16);
    gemm256_wmma<1><<<gg, 32, 0, stream>>>(hn,  Wq, tmp);  // tmp = hn @ Wq^T
    gemm256_wmma<0><<<gg, 32, 0, stream>>>(tmp, Wk, qt);   // qt  = tmp @ Wk

    const size_t smem_bytes = (size_t)(LSEQ * DIM + LSEQ + 8) * sizeof(float);
    attn_fused<<<NB, 256, smem_bytes, stream>>>(out_t, qt, pos_ind, rel_dt,
                                                pos_emb, rel_emb, Wrp, y);
}